// EMLChainWide_25022479467180
// MI455X (gfx1250) — compile-verified
//
#include <hip/hip_runtime.h>
#include <hip/hip_bf16.h>
#include <stdint.h>

// Reference: 8-level complex128 recurrence, elementwise over N=4M inputs.
//   fs = [1, x]; for i in 0..7:
//     w = softmax(logits[i,:,:i+2])            (88 scalar doubles total)
//     inp_l = sum w0k*fs[k]; inp_r = sum w1k*fs[k]
//     f = exp(clamp(Re(inp_l),±50))*cis(Im(inp_l)) - log(|inp_r|<1e-30 ? 1e-30 : inp_r)
//   out = fs[9]   (complex128 -> interleaved double2)
//
// Roofline: 80 MB traffic (3.4 us at 23.3 TB/s) vs ~7 GFLOP of *software* fp64
// transcendentals -> compute-bound on fp64 VALU. WMMA (f32 accumulate max) is
// numerically inapplicable at complex128 precision.
// CDNA5-specific paths: global_load_async_to_lds_b64 + s_wait_asynccnt
// (ASYNCcnt, CDNA5 ISA), global_prefetch_b8, NT cache hints.

#define NLEV 8
#define WSTRIDE 9                      // padded inner dim of logits / weights
#define NW (NLEV * 2 * WSTRIDE)        // 144 doubles
#define LOG_EPS (-69.07755278982137)   // log(1e-30)
#define EPS2 (1e-60)                   // (1e-30)^2

typedef double v2d __attribute__((ext_vector_type(2)));

// ---------------------------------------------------------------------------
// Kernel 1: compute the 16 softmax rows once (one per thread). Removes ~144
// fp64 exp() per element that a naive per-thread port would recompute.
// ---------------------------------------------------------------------------
__global__ void eml_weights_kernel(const double* __restrict__ logits,
                                   double* __restrict__ w) {
    int t = threadIdx.x;
    if (t >= NLEV * 2) return;
    int i = t >> 1;          // level
    int r = t & 1;           // row (l / r combo)
    int n = i + 2;           // valid width at this level
    const double* row = logits + (i * 2 + r) * WSTRIDE;
    double m = row[0];
    for (int k = 1; k < n; ++k) m = fmax(m, row[k]);
    double e[WSTRIDE];
    double sum = 0.0;
    for (int k = 0; k < n; ++k) { e[k] = exp(row[k] - m); sum += e[k]; }
    double inv = 1.0 / sum;
    double* outp = w + (i * 2 + r) * WSTRIDE;
    for (int k = 0; k < WSTRIDE; ++k) outp[k] = (k < n) ? e[k] * inv : 0.0;
}

// ---------------------------------------------------------------------------
// Kernel 2: the elementwise chain. Weights staged to LDS via the CDNA5
// async-to-LDS DMA path, then broadcast-read with ds_load_b128 pairs.
// ---------------------------------------------------------------------------
__global__ __launch_bounds__(256) void eml_chain_kernel(
    const float* __restrict__ x,
    const double* __restrict__ w,      // 144 doubles in d_ws
    v2d* __restrict__ out,
    int n) {
    __shared__ double ws[NW];

    const int tid = threadIdx.x;

    // Stage weight table into LDS with gfx1250 async load (ASYNCcnt path).
    if (tid < NW) {
        uint32_t lds_addr = (uint32_t)(uintptr_t)(&ws[tid]);   // low 32b = LDS offset
        uint32_t voff     = (uint32_t)(tid * 8);
        asm volatile("global_load_async_to_lds_b64 %0, %1, %2"
                     :
                     : "v"(lds_addr), "v"(voff), "s"(w)
                     : "memory");
        asm volatile("s_wait_asynccnt 0x0" ::: "memory");
    }
    __syncthreads();

    int idx = blockIdx.x * blockDim.x + tid;
    const int stride = gridDim.x * blockDim.x;

    for (; idx < n; idx += stride) {
        // Prefetch the next strided element of the x stream (global_prefetch_b8).
        if (idx + stride < n) __builtin_prefetch(&x[idx + stride], 0, 0);

        double fr[NLEV + 2];
        double fi[NLEV + 2];
        fr[0] = 1.0;
        fi[0] = 0.0;
        // NT load: x is a pure stream, read exactly once.
        fr[1] = (double)__builtin_nontemporal_load(&x[idx]);
        fi[1] = 0.0;

#pragma unroll
        for (int i = 0; i < NLEV; ++i) {
            double lr = 0.0, li = 0.0, rr = 0.0, ri = 0.0;
#pragma unroll
            for (int k = 0; k < i + 2; ++k) {
                const double w0 = ws[(i * 2 + 0) * WSTRIDE + k];
                const double w1 = ws[(i * 2 + 1) * WSTRIDE + k];
                lr = fma(w0, fr[k], lr);
                li = fma(w0, fi[k], li);
                rr = fma(w1, fr[k], rr);
                ri = fma(w1, fi[k], ri);
            }
            // safe_exp(inp_l): clamp real to +-50, cis on imag
            double er = fmin(fmax(lr, -50.0), 50.0);
            double ex = exp(er);
            double s, c;
            sincos(li, &s, &c);
            double exr = ex * c;
            double exi = ex * s;
            // safe_log(inp_r): |z|^2 test avoids the sqrt entirely
            double m2 = fma(rr, rr, ri * ri);
            bool small = (m2 < EPS2);
            double lgr = small ? LOG_EPS : 0.5 * log(m2);
            double lgi = small ? 0.0 : atan2(ri, rr);

            fr[i + 2] = exr - lgr;
            fi[i + 2] = exi - lgi;
        }

        v2d res;
        res.x = fr[NLEV + 1];
        res.y = fi[NLEV + 1];
        // NT 16B store (global_store_b128, TH=NT): output never re-read.
        __builtin_nontemporal_store(res, &out[idx]);
    }
}

extern "C" void kernel_launch(void* const* d_in, const int* in_sizes, int n_in,
                              void* d_out, int out_size, void* d_ws, size_t ws_size,
                              hipStream_t stream) {
    const float*  x      = (const float*)d_in[0];   // [N] float32
    const double* logits = (const double*)d_in[1];  // [8,2,9] float64
    double*       wtab   = (double*)d_ws;           // 144 doubles scratch
    v2d*          out    = (v2d*)d_out;             // complex128 interleaved

    const int N = in_sizes[0];

    // 1) softmax weights once (16 rows, one wave)
    eml_weights_kernel<<<1, 32, 0, stream>>>(logits, wtab);

    // 2) elementwise chain: 2 elements/thread via grid-stride; occupancy is
    //    the main latency-hiding lever for the long fp64 dependency chains.
    const int threads = 256;
    const int elems_per_thread = 2;
    int blocks = (N + threads * elems_per_thread - 1) / (threads * elems_per_thread);
    if (blocks < 1) blocks = 1;
    eml_chain_kernel<<<blocks, threads, 0, stream>>>(x, wtab, out, N);
}